// TverskyAttentionShared_35493609734446
// MI455X (gfx1250) — compile-verified
//
#include <hip/hip_runtime.h>
#include <hip/hip_bf16.h>
#include <math.h>

// ---------------------------------------------------------------------------
// TverskyAttentionShared for MI455X (gfx1250), f32 end-to-end.
// Dense contractions -> V_WMMA_F32_16X16X4_F32 (exact f32 path).
// Bulk global->LDS staging -> GLOBAL_LOAD_ASYNC_TO_LDS_B128 (ASYNCcnt).
// ---------------------------------------------------------------------------

typedef __attribute__((ext_vector_type(2))) float v2f;
typedef __attribute__((ext_vector_type(8))) float v8f;

#define DIM   512
#define EPS   1e-8f

#define WMMA_F32(a, b, c)                                                     \
  __builtin_amdgcn_wmma_f32_16x16x4_f32(false, (a), false, (b), (short)0,     \
                                        (c), false, false)

// Async copy of 16 bytes global -> LDS, tracked on ASYNCcnt (no VGPR data).
// GV mode: 64-bit global address in a VGPR pair, LDS byte address in VDST.
__device__ __forceinline__ void async_ld_b128(const float* g, float* l) {
  unsigned lds = (unsigned)(unsigned long long)l;   // low 32 bits = LDS offset
  asm volatile("global_load_async_to_lds_b128 %0, %1, off"
               :: "v"(lds), "v"(g) : "memory");
}
__device__ __forceinline__ void wait_async0() {
  asm volatile("s_wait_asynccnt 0x0" ::: "memory");
}

// ===========================================================================
// Kernel 1: C = relu(A @ B), all [512,512] row-major f32.
// Block = 8 waves = 256 thr. Block tile M=128 x N=64: wave w owns rows
// w*16..w*16+15, all 8 waves share one LDS-staged B chunk (64k x 64n,
// row stride 68 floats -> bank-conflict free, 16B aligned). Each wave keeps
// 4 accumulators (N sub-tiles), reusing its A fragment 4x per k-step.
// ===========================================================================
__global__ __launch_bounds__(256) void gemm_relu_512(
    const float* __restrict__ A, const float* __restrict__ B,
    float* __restrict__ C) {
  __shared__ __align__(16) float bs[64 * 68];

  const int t    = threadIdx.x;
  const int lane = t & 31;
  const int wave = t >> 5;
  const int half = lane >> 4;
  const int l16  = lane & 15;
  const int tm   = blockIdx.y * 128 + wave * 16;
  const int tn   = blockIdx.x * 64;

  const float* __restrict__ arow = A + (tm + l16) * DIM;

  v8f acc0 = {}, acc1 = {}, acc2 = {}, acc3 = {};

  for (int k0 = 0; k0 < DIM; k0 += 64) {
    __syncthreads();
    // stage B[k0..k0+63][tn..tn+63] -> LDS via async b128 (4 floats/thread/it)
#pragma unroll
    for (int i = 0; i < 4; ++i) {
      const int flat = (i * 256 + t) * 4;            // 0..4095
      const int r = flat >> 6, c = flat & 63;
      async_ld_b128(&B[(k0 + r) * DIM + tn + c], &bs[r * 68 + c]);
    }
    __builtin_prefetch(arow + k0 + 64, 0, 3);        // global_prefetch_b8
    wait_async0();
    __syncthreads();

#pragma unroll 4
    for (int kk = 0; kk < 64; kk += 4) {
      const int ka = kk + half * 2;
      v2f a;
      a.x = arow[k0 + ka];
      a.y = arow[k0 + ka + 1];
      const float* __restrict__ b0 = bs + ka * 68 + l16;
      const float* __restrict__ b1 = bs + (ka + 1) * 68 + l16;
      v2f b;
      b.x = b0[0];  b.y = b1[0];  acc0 = WMMA_F32(a, b, acc0);
      b.x = b0[16]; b.y = b1[16]; acc1 = WMMA_F32(a, b, acc1);
      b.x = b0[32]; b.y = b1[32]; acc2 = WMMA_F32(a, b, acc2);
      b.x = b0[48]; b.y = b1[48]; acc3 = WMMA_F32(a, b, acc3);
    }
  }

  const int row0 = tm + half * 8;
#pragma unroll
  for (int r = 0; r < 8; ++r) {
    float* __restrict__ crow = C + (row0 + r) * DIM + tn + l16;
    float v;
    v = acc0[r]; crow[0]  = v > 0.f ? v : 0.f;
    v = acc1[r]; crow[16] = v > 0.f ? v : 0.f;
    v = acc2[r]; crow[32] = v > 0.f ? v : 0.f;
    v = acc3[r]; crow[48] = v > 0.f ? v : 0.f;
  }
}

// ===========================================================================
// Kernel 2: Tversky combine (min-plus contraction; no tensor-core mapping).
//   common[n,m] = sum_d min(xf[n,d], pf[m,d])
//   out[n,m]    = g*c / (g*c + |a|*(sx[n]-c) + |b|*(sp[m]-c) + eps) + bias[m]
// 32x32 tile per block, d streamed in 128-wide LDS chunks (stride 132,
// async-b128 staged). Row sums fused into the same pass.
// ===========================================================================
__global__ __launch_bounds__(256) void tversky_512(
    const float* __restrict__ XF, const float* __restrict__ PF,
    const float* __restrict__ bias,
    const float* __restrict__ pa, const float* __restrict__ pb,
    const float* __restrict__ pg,
    float* __restrict__ OUT) {
  __shared__ __align__(16) float xs[32 * 132];
  __shared__ __align__(16) float ps[32 * 132];

  const int t  = threadIdx.x;
  const int n0 = blockIdx.y * 32;
  const int m0 = blockIdx.x * 32;
  const int ln = t >> 3;            // local n: 0..31
  const int mg = (t & 7) * 4;       // local m base: 0,4,..,28

  float c0 = 0.f, c1 = 0.f, c2 = 0.f, c3 = 0.f;
  float sx = 0.f;
  float sp0 = 0.f, sp1 = 0.f, sp2 = 0.f, sp3 = 0.f;

  for (int d0 = 0; d0 < DIM; d0 += 128) {
    __syncthreads();
#pragma unroll
    for (int i = 0; i < 4; ++i) {   // 32*128 floats / (256 thr * 4) = 4 iters
      const int flat = (i * 256 + t) * 4;
      const int r = flat >> 7, col = flat & 127;
      async_ld_b128(&XF[(n0 + r) * DIM + d0 + col], &xs[r * 132 + col]);
      async_ld_b128(&PF[(m0 + r) * DIM + d0 + col], &ps[r * 132 + col]);
    }
    wait_async0();
    __syncthreads();

    const float* __restrict__ xr = xs + ln * 132;
    const float* __restrict__ p0 = ps + (mg + 0) * 132;
    const float* __restrict__ p1 = ps + (mg + 1) * 132;
    const float* __restrict__ p2 = ps + (mg + 2) * 132;
    const float* __restrict__ p3 = ps + (mg + 3) * 132;

#pragma unroll 4
    for (int d = 0; d < 128; ++d) {
      const float xv = xr[d];
      sx += xv;
      const float a0 = p0[d], a1 = p1[d], a2 = p2[d], a3 = p3[d];
      sp0 += a0; sp1 += a1; sp2 += a2; sp3 += a3;
      c0 += fminf(xv, a0);
      c1 += fminf(xv, a1);
      c2 += fminf(xv, a2);
      c3 += fminf(xv, a3);
    }
  }

  const float aa = fabsf(pa[0]);
  const float bb = fabsf(pb[0]);
  const float gg = pg[0];
  const int n = n0 + ln;
  const int m = m0 + mg;

  float num, den;
  num = gg * c0; den = num + aa * (sx - c0) + bb * (sp0 - c0) + EPS;
  OUT[n * DIM + m + 0] = num / den + bias[m + 0];
  num = gg * c1; den = num + aa * (sx - c1) + bb * (sp1 - c1) + EPS;
  OUT[n * DIM + m + 1] = num / den + bias[m + 1];
  num = gg * c2; den = num + aa * (sx - c2) + bb * (sp2 - c2) + EPS;
  OUT[n * DIM + m + 2] = num / den + bias[m + 2];
  num = gg * c3; den = num + aa * (sx - c3) + bb * (sp3 - c3) + EPS;
  OUT[n * DIM + m + 3] = num / den + bias[m + 3];
}

// ===========================================================================
// Kernel 3: per-head attention, flash-style per 16-q-row tile.
// Block = 4 waves (128 thr), grid = (32 q-tiles, 8 heads).
//  phase1: scores (16x512) = Q_tile @ K^T / 8 via WMMA f32 -> LDS (stride 516)
//  phase2: row softmax fully in LDS
//  phase3: ctx (16x64) = attn @ V_head via WMMA f32 -> global (merged heads)
// ===========================================================================
__global__ __launch_bounds__(128) void attn_512(
    const float* __restrict__ Q, const float* __restrict__ K,
    const float* __restrict__ V, float* __restrict__ O) {
  __shared__ __align__(16) float sc[16 * 516];   // 516 % 64 = 4: no conflicts
  __shared__ float red[16][8];
  __shared__ float rowstat[16];

  const int h     = blockIdx.y;
  const int q0    = blockIdx.x * 16;
  const int t     = threadIdx.x;
  const int lane  = t & 31;
  const int wave  = t >> 5;
  const int half  = lane >> 4;
  const int l16   = lane & 15;
  const int hbase = h * 64;

  // ---- phase 1: scores -> LDS ----
  const float* __restrict__ qrow = Q + (q0 + l16) * DIM + hbase;
  for (int kt = wave; kt < 32; kt += 4) {
    const float* __restrict__ krow = K + (kt * 16 + l16) * DIM + hbase;
    v8f acc = {};
#pragma unroll
    for (int kk = 0; kk < 64; kk += 4) {
      const int ka = kk + half * 2;
      v2f a, b;
      a.x = qrow[ka]; a.y = qrow[ka + 1];
      b.x = krow[ka]; b.y = krow[ka + 1];   // B = K^T: N lane = key row
      acc = WMMA_F32(a, b, acc);
    }
    const int r0 = half * 8;
#pragma unroll
    for (int r = 0; r < 8; ++r)
      sc[(r0 + r) * 516 + kt * 16 + l16] = acc[r] * 0.125f;  // 1/sqrt(64)
  }
  __syncthreads();

  // ---- phase 2: softmax over each of the 16 rows (8 threads/row) ----
  const int row = t >> 3;
  const int seg = t & 7;
  float* __restrict__ srow = sc + row * 516 + seg * 64;

  float mx = -3.4e38f;
#pragma unroll 8
  for (int i = 0; i < 64; ++i) mx = fmaxf(mx, srow[i]);
  red[row][seg] = mx;
  __syncthreads();
  if (seg == 0) {
    float mm = red[row][0];
#pragma unroll
    for (int i = 1; i < 8; ++i) mm = fmaxf(mm, red[row][i]);
    rowstat[row] = mm;
  }
  __syncthreads();
  const float rm = rowstat[row];
  float sum = 0.f;
#pragma unroll 8
  for (int i = 0; i < 64; ++i) {
    const float e = __expf(srow[i] - rm);
    srow[i] = e;
    sum += e;
  }
  red[row][seg] = sum;
  __syncthreads();
  if (seg == 0) {
    float ss = 0.f;
#pragma unroll
    for (int i = 0; i < 8; ++i) ss += red[row][i];
    rowstat[row] = ss;
  }
  __syncthreads();
  const float inv = 1.0f / rowstat[row];
#pragma unroll 8
  for (int i = 0; i < 64; ++i) srow[i] *= inv;
  __syncthreads();

  // ---- phase 3: ctx = attn @ V_head ; wave w owns head-dim cols w*16.. ----
  {
    v8f acc = {};
    const int ncol = hbase + wave * 16 + l16;
    for (int kk = 0; kk < DIM; kk += 4) {
      const int ka = kk + half * 2;
      v2f a, b;
      a.x = sc[l16 * 516 + ka];          // A from LDS (ds loads)
      a.y = sc[l16 * 516 + ka + 1];
      b.x = V[ka * DIM + ncol];
      b.y = V[(ka + 1) * DIM + ncol];
      acc = WMMA_F32(a, b, acc);
    }
    const int r0 = half * 8;
#pragma unroll
    for (int r = 0; r < 8; ++r)
      O[(q0 + r0 + r) * DIM + ncol] = acc[r];   // merged-head [N, D] layout
  }
}

// ===========================================================================
// Host-side orchestration. Workspace (f32 slots of 512*512):
//   0: xf (-> ctx)  1: pfq  2: pfk  3: pfv  4: pfo
//   5: q (-> ctxf)  6: k    7: v          Total scratch = 8 MB (L2-resident).
// ===========================================================================
extern "C" void kernel_launch(void* const* d_in, const int* in_sizes, int n_in,
                              void* d_out, int out_size, void* d_ws,
                              size_t ws_size, hipStream_t stream) {
  const float* x        = (const float*)d_in[0];   // [1,512,512] -> [512,512]
  const float* features = (const float*)d_in[1];
  const float* alpha    = (const float*)d_in[2];
  const float* beta     = (const float*)d_in[3];
  const float* gamma    = (const float*)d_in[4];
  const float* proto_q  = (const float*)d_in[5];
  const float* bias_q   = (const float*)d_in[6];
  const float* proto_k  = (const float*)d_in[7];
  const float* bias_k   = (const float*)d_in[8];
  const float* proto_v  = (const float*)d_in[9];
  const float* bias_v   = (const float*)d_in[10];
  const float* proto_o  = (const float*)d_in[11];
  const float* bias_o   = (const float*)d_in[12];
  float* out = (float*)d_out;

  float* ws = (float*)d_ws;
  const size_t SLOT = (size_t)DIM * DIM;
  float* xf   = ws + 0 * SLOT;
  float* pfq  = ws + 1 * SLOT;
  float* pfk  = ws + 2 * SLOT;
  float* pfv  = ws + 3 * SLOT;
  float* pfo  = ws + 4 * SLOT;
  float* qm   = ws + 5 * SLOT;
  float* km   = ws + 6 * SLOT;
  float* vm   = ws + 7 * SLOT;
  float* ctx  = xf;   // xf dead after q/k/v
  float* ctxf = qm;   // qm dead after attention

  const dim3 gGemm(8, 4), bGemm(256);         // N-groups of 64 x M-groups 128
  const dim3 gTv(16, 16), bTv(256);           // 32x32 output tiles
  const dim3 gAt(32, 8), bAt(128);            // (q-tile, head)

  gemm_relu_512<<<gGemm, bGemm, 0, stream>>>(x,       features, xf);
  gemm_relu_512<<<gGemm, bGemm, 0, stream>>>(proto_q, features, pfq);
  gemm_relu_512<<<gGemm, bGemm, 0, stream>>>(proto_k, features, pfk);
  gemm_relu_512<<<gGemm, bGemm, 0, stream>>>(proto_v, features, pfv);
  gemm_relu_512<<<gGemm, bGemm, 0, stream>>>(proto_o, features, pfo);

  tversky_512<<<gTv, bTv, 0, stream>>>(xf, pfq, bias_q, alpha, beta, gamma, qm);
  tversky_512<<<gTv, bTv, 0, stream>>>(xf, pfk, bias_k, alpha, beta, gamma, km);
  tversky_512<<<gTv, bTv, 0, stream>>>(xf, pfv, bias_v, alpha, beta, gamma, vm);

  attn_512<<<gAt, bAt, 0, stream>>>(qm, km, vm, ctx);

  gemm_relu_512<<<gGemm, bGemm, 0, stream>>>(ctx, features, ctxf);
  tversky_512<<<gTv, bTv, 0, stream>>>(ctxf, pfo, bias_o, alpha, beta, gamma,
                                       out);
}